// RadonForward_2817498546706
// MI455X (gfx1250) — compile-verified
//
#include <hip/hip_runtime.h>

// Radon transform, MI455X (gfx1250).
// B=4, C=1, H=W=256, A=180, P=256, T=ceil(sqrt(2)*256)=363.
// One workgroup per (batch, 2-angle tile): 360 WGs. The whole 256KB image is
// DMA'd into LDS by ONE tensor_load_to_lds (TDM) with hardware row padding
// (row stride 257 floats -> bank = (x+y) mod 64, conflict-free for both
// horizontal and vertical ray walks). 512 threads = 16 waves (4/SIMD for LDS
// latency hiding); thread (aw, p) owns angle atile*2+aw, position p, and
// privately accumulates its 363 bilinear ray samples from LDS.

#define HH 256
#define WW 256
#define AA 180
#define PP 256
#define TT 363
#define ANG_PER_WG 2
#define LDS_STRIDE 257                    // 256 data + 1 pad DWORD per row
#define LDS_FLOATS (HH * LDS_STRIDE)      // 65792 floats = 263168 bytes

typedef unsigned int u32x4 __attribute__((ext_vector_type(4)));
typedef int          i32x8 __attribute__((ext_vector_type(8)));
typedef int          i32x4 __attribute__((ext_vector_type(4)));

extern __shared__ float tile[];           // dynamic LDS, offset 0 (no static LDS)

__device__ __forceinline__ float fetch_lds(int yy, int xx) {
    // valid iff 0 <= xx < 256 && 0 <= yy < 256 (unsigned-compare trick)
    bool v = ((unsigned)xx < (unsigned)WW) & ((unsigned)yy < (unsigned)HH);
    int xc = min(max(xx, 0), WW - 1);
    int yc = min(max(yy, 0), HH - 1);
    float val = tile[yc * LDS_STRIDE + xc];
    return v ? val : 0.0f;
}

__global__ void __launch_bounds__(512, 1)
radon_fwd_kernel(const float* __restrict__ x,
                 const float* __restrict__ thetas,
                 const float* __restrict__ positions,
                 float* __restrict__ out) {
    const int tid   = threadIdx.x;
    const int p     = tid & (PP - 1);      // position index, 0..255
    const int aw    = tid >> 8;            // which of the 2 angles, 0..1
    const int atile = blockIdx.x;          // angle tile, 0..89
    const int b     = blockIdx.y;          // batch, 0..3

    // ---- Stage full image for this batch into LDS via the Tensor Data Mover.
    // One DMA op per workgroup, issued by wave 0 only (TDM ignores EXEC).
    if (tid < 32) {
        unsigned long long ga =
            (unsigned long long)(const void*)(x + (size_t)b * HH * WW);

        // D# group 0 (CDNA5 ISA)
        u32x4 g0;
        g0.x = 1u;                                        // count=1, user mode
        g0.y = 0u;                                        // lds_addr = 0
        g0.z = (unsigned)(ga & 0xFFFFFFFFull);            // global_addr[31:0]
        g0.w = (unsigned)((ga >> 32) & 0x1FFFFFFull)      // global_addr[56:32]
             | (2u << 30);                                // type = 2 ("image")

        // D# group 1 (§8.4): data_size=4B (2), pad_enable, pad_interval code 7
        // (=256 DWORDs, i.e. one row), pad_amount code 0 (=1 DWORD).
        i32x8 g1;
        g1[0] = (2 << 16) | (1 << 20) | (7 << 22);        // wg_mask=0 | ds=4B | pad
        g1[1] = (int)(256u << 16);                        // tensor_dim0 lo16 = 256
        g1[2] = (int)(256u << 16);                        // dim0 hi=0 | tensor_dim1 lo16 = 256
        g1[3] = (int)(256u << 16);                        // dim1 hi=0 | tile_dim0 = 256
        g1[4] = 256;                                      // tile_dim1 = 256 | tile_dim2 = 0
        g1[5] = 256;                                      // tensor_dim0_stride lo32 = 256
        g1[6] = 0;                                        // stride0 hi | stride1 lo16
        g1[7] = 1;                                        // tensor_dim1_stride = 65536 (hi bits)

        i32x4 gz  = {0, 0, 0, 0};                         // groups 2/3: 2-D tile
        i32x8 gz8 = {0, 0, 0, 0, 0, 0, 0, 0};            // trailing group (6-arg form)

        __builtin_amdgcn_tensor_load_to_lds(g0, g1, gz, gz, gz8, 0);
        __builtin_amdgcn_s_wait_tensorcnt(0);
    }
    __syncthreads();

    // ---- Compute: this thread owns (angle a, position p).
    const int   a  = atile * ANG_PER_WG + aw;
    const float s  = positions[p];                        // p - 127.5
    const float cx = 0.5f * (float)(WW - 1);              // 127.5
    const float cy = 0.5f * (float)(HH - 1);              // 127.5
    const float tc = 0.5f * (float)(TT - 1);              // 181.0

    const float th = thetas[a];
    const float cs = __cosf(th);
    const float sn = __sinf(th);

    const float bx = cx + s * cs;                         // ray base point
    const float by = cy + s * sn;

    float acc = 0.0f;
    #pragma unroll 4
    for (int t = 0; t < TT; ++t) {
        const float ttv = (float)t - tc;
        const float px = bx - ttv * sn;
        const float py = by + ttv * cs;
        const float fx = floorf(px);
        const float fy = floorf(py);
        const float wx = px - fx;
        const float wy = py - fy;
        const int ix = (int)fx;
        const int iy = (int)fy;

        const float v00 = fetch_lds(iy,     ix);
        const float v10 = fetch_lds(iy,     ix + 1);
        const float v01 = fetch_lds(iy + 1, ix);
        const float v11 = fetch_lds(iy + 1, ix + 1);

        const float ox = 1.0f - wx;
        const float oy = 1.0f - wy;
        acc += v00 * (ox * oy) + v10 * (wx * oy)
             + v01 * (ox * wy) + v11 * (wx * wy);
    }
    out[((size_t)b * AA + a) * PP + p] = acc;
}

extern "C" void kernel_launch(void* const* d_in, const int* in_sizes, int n_in,
                              void* d_out, int out_size, void* d_ws, size_t ws_size,
                              hipStream_t stream) {
    (void)in_sizes; (void)n_in; (void)out_size; (void)d_ws; (void)ws_size;
    const float* x         = (const float*)d_in[0];   // [4,1,256,256] f32
    const float* thetas    = (const float*)d_in[1];   // [180] f32
    const float* positions = (const float*)d_in[2];   // [256] f32
    float* out             = (float*)d_out;           // [4,1,180,256] f32

    const size_t shmem = (size_t)LDS_FLOATS * sizeof(float); // 263168 B <= 320KB/WGP
    // Opt in to >64KB dynamic LDS (no-op if already allowed). Host-side call,
    // not stream-ordered: safe under graph capture.
    (void)hipFuncSetAttribute((const void*)radon_fwd_kernel,
                              hipFuncAttributeMaxDynamicSharedMemorySize,
                              (int)shmem);

    dim3 grid(AA / ANG_PER_WG, 4, 1);   // 90 angle tiles x 4 batches = 360 WGs
    dim3 block(512, 1, 1);              // 16 waves of 32 (4 per SIMD32)
    radon_fwd_kernel<<<grid, block, shmem, stream>>>(x, thetas, positions, out);
}